// HoloLinear_14946486190460
// MI455X (gfx1250) — compile-verified
//
#include <hip/hip_runtime.h>
#include <math.h>

typedef __attribute__((ext_vector_type(2))) float v2f;
typedef __attribute__((ext_vector_type(8))) float v8f;

#define TOKENS 16384
#define IN_F   1024
#define OUT_F  4096
#define HARM   32

// ---------------------------------------------------------------------------
// Stage 1: w_real[o][h] = amp[o][h] * cos(phase[o][h])          (4096 x 32)
// ---------------------------------------------------------------------------
__global__ __launch_bounds__(256) void wreal_kernel(const float* __restrict__ phase,
                                                    const float* __restrict__ amp,
                                                    float* __restrict__ wr) {
    int i = blockIdx.x * blockDim.x + threadIdx.x;
    if (i < OUT_F * HARM) wr[i] = amp[i] * cosf(phase[i]);
}

// ---------------------------------------------------------------------------
// Stage 2: resonance = x @ basis^T   (16384 x 1024) x (1024 x 32) -> 16384 x 32
// One wave per 16-token tile. Two N-tiles (harmonics 0-15, 16-31) share the
// A fragment so each element of x is loaded exactly once (64 MB HBM read).
// f32 WMMA 16x16x4, K swept 4 at a time (256 steps, 2 WMMA each).
// Fragment layouts per CDNA5 ISA 7.12.2:
//   A: lane(half,m): a[j] = A[m][k0 + j + 2*half]
//   B: lane(half,m): b[j] = B[k0 + j + 2*half][m]
//   D: vgpr g:       D[g + 8*half][m]
// ---------------------------------------------------------------------------
__global__ __launch_bounds__(32) void resonance_kernel(const float* __restrict__ x,
                                                       const float* __restrict__ basis,
                                                       float* __restrict__ res) {
    const int lane = threadIdx.x;
    const int m    = lane & 15;
    const int half = lane >> 4;
    const int t0   = blockIdx.x * 16;

    const float* ax  = x     + (size_t)(t0 + m) * IN_F + 2 * half;
    const float* bp0 = basis + (size_t)(m)      * IN_F + 2 * half;   // harmonics 0..15
    const float* bp1 = basis + (size_t)(16 + m) * IN_F + 2 * half;   // harmonics 16..31

    v8f acc0 = {};
    v8f acc1 = {};

#pragma unroll 4
    for (int k = 0; k < IN_F; k += 4) {
        v2f a  = *(const v2f*)(ax  + k);
        v2f b0 = *(const v2f*)(bp0 + k);
        v2f b1 = *(const v2f*)(bp1 + k);
        acc0 = __builtin_amdgcn_wmma_f32_16x16x4_f32(false, a, false, b0,
                                                     (short)0, acc0, false, false);
        acc1 = __builtin_amdgcn_wmma_f32_16x16x4_f32(false, a, false, b1,
                                                     (short)0, acc1, false, false);
    }

#pragma unroll
    for (int g = 0; g < 8; ++g) {
        const int row = t0 + g + 8 * half;
        res[(size_t)row * HARM + m]      = acc0[g];
        res[(size_t)row * HARM + 16 + m] = acc1[g];
    }
}

// ---------------------------------------------------------------------------
// Stage 3: out = resonance @ w_real^T  (16384 x 32) x (32 x 4096) -> 16384 x 4096
// K = 32 (8 WMMA k-steps). Each wave loads its A fragments once, then emits
// 4 adjacent 16x16 output tiles. resonance (2 MB) and w_real (512 KB) are
// L2-resident; HBM traffic is the 256 MB output write.
// ---------------------------------------------------------------------------
#define NWAVE 8
#define OT_PER_WAVE 4

__global__ __launch_bounds__(256) void out_kernel(const float* __restrict__ res,
                                                  const float* __restrict__ wr,
                                                  float* __restrict__ out) {
    const int lane = threadIdx.x & 31;
    const int wave = threadIdx.x >> 5;
    const int m    = lane & 15;
    const int half = lane >> 4;
    const int t0   = blockIdx.x * 16;

    // A fragments for the whole K=32 sweep (16 tokens x 32 harmonics)
    const float* ap = res + (size_t)(t0 + m) * HARM + 2 * half;
    v2f a[8];
#pragma unroll
    for (int s = 0; s < 8; ++s) a[s] = *(const v2f*)(ap + 4 * s);

    const int otbase = (blockIdx.y * NWAVE + wave) * OT_PER_WAVE;

#pragma unroll
    for (int ot = 0; ot < OT_PER_WAVE; ++ot) {
        const int o0 = (otbase + ot) * 16;
        const float* bp = wr + (size_t)(o0 + m) * HARM + 2 * half;

        v8f acc = {};
#pragma unroll
        for (int s = 0; s < 8; ++s) {
            v2f b = *(const v2f*)(bp + 4 * s);
            acc = __builtin_amdgcn_wmma_f32_16x16x4_f32(false, a[s], false, b,
                                                        (short)0, acc, false, false);
        }

#pragma unroll
        for (int g = 0; g < 8; ++g) {
            out[(size_t)(t0 + g + 8 * half) * OUT_F + o0 + m] = acc[g];
        }
    }
}

// ---------------------------------------------------------------------------
extern "C" void kernel_launch(void* const* d_in, const int* in_sizes, int n_in,
                              void* d_out, int out_size, void* d_ws, size_t ws_size,
                              hipStream_t stream) {
    const float* x     = (const float*)d_in[0];   // [16384, 1024]
    const float* basis = (const float*)d_in[1];   // [32, 1024]
    const float* phase = (const float*)d_in[2];   // [4096, 32]
    const float* amp   = (const float*)d_in[3];   // [4096, 32]
    float* out = (float*)d_out;                   // [16384, 4096]

    float* res = (float*)d_ws;                    // 16384*32 f32 = 2 MB
    float* wr  = res + (size_t)TOKENS * HARM;     // 4096*32 f32  = 512 KB

    wreal_kernel<<<(OUT_F * HARM + 255) / 256, 256, 0, stream>>>(phase, amp, wr);
    resonance_kernel<<<TOKENS / 16, 32, 0, stream>>>(x, basis, res);
    out_kernel<<<dim3(TOKENS / 16, OUT_F / (16 * NWAVE * OT_PER_WAVE)), 256, 0, stream>>>(res, wr, out);
}